// projectionLoss_30451318129159
// MI455X (gfx1250) — compile-verified
//
#include <hip/hip_runtime.h>

typedef float v2f __attribute__((ext_vector_type(2)));
typedef float v8f __attribute__((ext_vector_type(8)));

#define GRIDSZ 64
#define NPIX   4096            // 64*64
#define NPTS   30000
#define NBATCH 16
#define SPLIT  8               // blocks per image (30000/8 = 3750 pts each)
#define NIMG   64              // 2 rot * 2 clouds * 16 batches
#define PTS_PER_SLICE (NPTS / SPLIT)                 // 3750
#define TILE   256                                   // points staged per block-tile
#define TILE_DW (TILE * 3)                           // 768 dwords = 3 KB
#define NTILES ((PTS_PER_SLICE + TILE - 1) / TILE)   // 15

// ---------------------------------------------------------------- init
__global__ void init_kernel(float* __restrict__ imgs, float* __restrict__ out) {
  int idx = blockIdx.x * blockDim.x + threadIdx.x;
  for (int t = idx; t < NIMG * NPIX; t += gridDim.x * blockDim.x) imgs[t] = 0.0f;
  if (idx == 0) out[0] = 0.0f;
}

// Async copy of one point tile (768 dwords) into LDS; always issues exactly
// 3 global_load_async_to_lds_b32 per wave (addresses clamped in-bounds) so
// ASYNCcnt advances by a constant 3 per tile -> clean s_wait_asynccnt 0x3
// double buffering.
__device__ __forceinline__ void prefetch_tile(const float* __restrict__ gbase,
                                              int tileDwStart, int maxDw,
                                              float* ldsTile) {
  const int t = threadIdx.x;
#pragma unroll
  for (int k = 0; k < 3; ++k) {
    int d = t + k * TILE;                       // 0..767
    int g = tileDwStart + d;
    g = (g < maxDw) ? g : (maxDw - 1);          // clamp tail, stay in-bounds
    unsigned int loff = (unsigned int)(unsigned long long)(uintptr_t)(ldsTile + d);
    unsigned long long ga = (unsigned long long)(uintptr_t)(gbase + g);
    asm volatile("global_load_async_to_lds_b32 %0, %1, off"
                 :: "v"(loff), "v"(ga) : "memory");
  }
}

// ---------------------------------------------------------------- splat
// grid = NIMG * SPLIT blocks. LDS-privatized 64x64 image, double-buffered
// async point staging, rotation via V_WMMA_F32_16X16X4_F32, splat via
// ds_add_f32 (wave-uniform fast path: 25 unconditional taps), merge via
// global_atomic_add_f32.
__global__ __launch_bounds__(256) void splat_kernel(const float* __restrict__ pc1,
                                                    const float* __restrict__ pc2,
                                                    const float* __restrict__ rot,
                                                    float* __restrict__ imgs) {
  __shared__ float img[NPIX];
  __shared__ float ptile[2][TILE_DW];
  __shared__ float zeroSlot;              // zeroed LDS dword for B-matrix K1/K3 padding reads

  const int blk   = blockIdx.x;
  const int image = blk >> 3;          // [0,64)
  const int slice = blk & (SPLIT - 1); // [0,8)
  const int r  = image >> 5;           // rotation index
  const int cl = (image >> 4) & 1;     // cloud index
  const int ba = image & 15;           // batch index

  const float* pts = (cl == 0 ? pc1 : pc2) + (size_t)ba * NPTS * 3;

  // ---- rotation matrix (uniform per block)
  const float t0 = rot[r * 3 + 0], t1 = rot[r * 3 + 1], t2 = rot[r * 3 + 2];
  const float s0 = sinf(t0), c0 = cosf(t0);
  const float s1 = sinf(t1), c1 = cosf(t1);
  const float s2 = sinf(t2), c2 = cosf(t2);
  const float R00 = c1 * c2, R01 = s0 * s1 * c2 - s2 * c0, R02 = s1 * c0 * c2 + s0 * s2;
  const float R10 = s2 * c1, R11 = s0 * s1 * s2 + c0 * c2, R12 = s1 * s2 * c0 - s0 * c2;
  const float R20 = -s1,     R21 = s0 * c1,                R22 = c0 * c1;

  // ---- 5x5 Gaussian stencil weights in VGPRs
  float w[25];
#pragma unroll
  for (int i = 0; i < 5; ++i)
#pragma unroll
    for (int j = 0; j < 5; ++j) {
      float di = (float)i - 5.0f, dj = (float)j - 5.0f;
      w[i * 5 + j] = expf(-sqrtf(di * di + dj * dj) * 2.5f); // 1/(2*0.2)
    }

  for (int t = threadIdx.x; t < NPIX; t += blockDim.x) img[t] = 0.0f;
  if (threadIdx.x == 0) zeroSlot = 0.0f;

  const int lane = threadIdx.x & 31;
  const int wid  = threadIdx.x >> 5;
  const int lo16 = lane & 15;

  // ---- WMMA A-matrix = R (16x4, K=3 zero-padded). ISA A layout:
  // lane m (0-15): vgpr0=A[m][0], vgpr1=A[m][1]; lane 16+m: vgpr0=A[m][2], vgpr1=A[m][3].
  v2f A; A.x = 0.0f; A.y = 0.0f;
  if (lane == 0)       { A.x = R00; A.y = R01; }
  else if (lane == 1)  { A.x = R10; A.y = R11; }
  else if (lane == 2)  { A.x = R20; A.y = R21; }
  else if (lane == 16) { A.x = R02; }
  else if (lane == 17) { A.x = R12; }
  else if (lane == 18) { A.x = R22; }

  const int pbase = slice * PTS_PER_SLICE;
  const int pend  = pbase + PTS_PER_SLICE;
  const int dwBase = pbase * 3;
  const int dwEnd  = pend * 3;

  prefetch_tile(pts, dwBase, dwEnd, &ptile[0][0]);

  for (int tk = 0; tk < NTILES; ++tk) {
    const int buf = tk & 1;
    const bool more = (tk + 1 < NTILES);
    if (more) prefetch_tile(pts, dwBase + (tk + 1) * TILE_DW, dwEnd, &ptile[buf ^ 1][0]);
    if (more) asm volatile("s_wait_asynccnt 0x3" ::: "memory");  // tile tk landed
    else      asm volatile("s_wait_asynccnt 0x0" ::: "memory");
    __syncthreads();   // all waves' async writes for this tile visible

    const float* tile = &ptile[buf][0];

    // B[k][n] = coord k of point n (ISA 4x16 B layout: vgpr0 = rows K0|K2,
    // vgpr1 = rows K1|K3(=0)). All loads unconditional: lanes>=16 read the
    // vgpr1 value from a zeroed LDS slot (address select, no exec juggling).
    const int l1 = wid * 32 + lo16;
    const int c0 = (lane < 16) ? 0 : 2;
    v2f B1, B2;
    B1.x = tile[l1 * 3 + c0];
    B1.y = *((lane < 16) ? &tile[l1 * 3 + 1] : &zeroSlot);
    B2.x = tile[(l1 + 16) * 3 + c0];
    B2.y = *((lane < 16) ? &tile[(l1 + 16) * 3 + 1] : &zeroSlot);

    v8f Cz = {0.f, 0.f, 0.f, 0.f, 0.f, 0.f, 0.f, 0.f};
    // D[m][n] = sum_k R[m][k]*pt_n[k] -> vgpr0/1/2 @ lanes0-15 = cam x/y/z of pt n=lane
    v8f D1 = __builtin_amdgcn_wmma_f32_16x16x4_f32(false, A, false, B1, (short)0, Cz, false, false);
    v8f D2 = __builtin_amdgcn_wmma_f32_16x16x4_f32(false, A, false, B2, (short)0, Cz, false, false);

    // lanes 16-31 take point p+lane from D2 lanes 0-15
    float cx2 = __shfl(D2[0], lo16, 32);
    float cy2 = __shfl(D2[1], lo16, 32);
    float cz2 = __shfl(D2[2], lo16, 32);
    float cx = (lane < 16) ? D1[0] : cx2;
    float cy = (lane < 16) ? D1[1] : cy2;
    float cz = (lane < 16) ? D1[2] : cz2;

    const float z = cz + 2.5f;
    const float s = 120.0f / z;            // one divide instead of two
    const float u = cx * s + 36.0f;
    const float v = cy * s + 36.0f;
    const int x4 = (int)rintf(u);          // round-half-even == jnp.round
    const int y4 = (int)rintf(v);
    const int myPt = pbase + tk * TILE + wid * 32 + lane;
    const bool ok = (myPt < pend) && ((unsigned)x4 <= 64u) && ((unsigned)y4 <= 64u);

    // ok implies px,py <= 62 (upper-bound checks redundant). Taps only clip
    // within 6 px of the low border; ~99% of lanes have x4>=6 && y4>=6 where
    // all 25 taps are valid -> wave-uniform fast path: 25 unconditional
    // ds_add_f32 off one base address with immediate offsets.
    const int base2 = (x4 - 6) * GRIDSZ + (y4 - 6);
    const bool fast = ok && (x4 >= 6) && (y4 >= 6);
    if (__all(fast)) {
      float* p = &img[base2];
#pragma unroll
      for (int i = 0; i < 5; ++i)
#pragma unroll
        for (int j = 0; j < 5; ++j)
          atomicAdd(p + (4 - i) * GRIDSZ + (4 - j), w[i * 5 + j]);   // ds_add_f32
    } else if (__any(ok)) {
#pragma unroll
      for (int i = 0; i < 5; ++i) {
        const bool vi = ok && (x4 >= i + 2);
#pragma unroll
        for (int j = 0; j < 5; ++j) {
          if (vi && (y4 >= j + 2))
            atomicAdd(&img[base2 + (4 - i) * GRIDSZ + (4 - j)], w[i * 5 + j]);
        }
      }
    }
    __syncthreads();   // everyone done with ptile[buf] before it is re-filled
  }
  __syncthreads();

  // merge partial LDS image into global accumulator (L2-resident, 1 MB total)
  float* gimg = imgs + (size_t)image * NPIX;
  for (int t = threadIdx.x; t < NPIX; t += blockDim.x) {
    float val = img[t];
    if (val != 0.0f) atomicAdd(&gimg[t], val);   // global_atomic_add_f32
  }
}

// ---------------------------------------------------------------- fused normalize + BCE
// 32 blocks: (rot, batch). Max-reduce both raw images, then BCE with on-the-fly
// normalization; one global atomic per block into the scalar loss.
__global__ __launch_bounds__(256) void norm_bce_kernel(const float* __restrict__ imgs,
                                                       float* __restrict__ out) {
  __shared__ float red[256];
  const int r = blockIdx.x >> 4, ba = blockIdx.x & 15;
  const float* g1 = imgs + (size_t)((r * 2 + 0) * NBATCH + ba) * NPIX;  // input cloud
  const float* g2 = imgs + (size_t)((r * 2 + 1) * NBATCH + ba) * NPIX;  // target cloud

  float m1 = 0.0f, m2 = 0.0f;
  for (int t = threadIdx.x; t < NPIX; t += blockDim.x) {
    m1 = fmaxf(m1, g1[t]);
    m2 = fmaxf(m2, g2[t]);
  }
  red[threadIdx.x] = m1;
  __syncthreads();
  for (int s = 128; s > 0; s >>= 1) {
    if (threadIdx.x < s) red[threadIdx.x] = fmaxf(red[threadIdx.x], red[threadIdx.x + s]);
    __syncthreads();
  }
  const float max1 = red[0];
  __syncthreads();
  red[threadIdx.x] = m2;
  __syncthreads();
  for (int s = 128; s > 0; s >>= 1) {
    if (threadIdx.x < s) red[threadIdx.x] = fmaxf(red[threadIdx.x], red[threadIdx.x + s]);
    __syncthreads();
  }
  const float max2 = red[0];
  __syncthreads();

  float acc = 0.0f;
  for (int t = threadIdx.x; t < NPIX; t += blockDim.x) {
    float p = g1[t] / max1;              // all-zero image -> 0/0 = NaN, same as ref
    float q = g2[t] / max2;
    float lp = fmaxf(logf(p), -100.0f);
    float lq = fmaxf(logf(1.0f - p), -100.0f);
    acc += q * lp + (1.0f - q) * lq;
  }
  red[threadIdx.x] = acc;
  __syncthreads();
  for (int s = 128; s > 0; s >>= 1) {
    if (threadIdx.x < s) red[threadIdx.x] += red[threadIdx.x + s];
    __syncthreads();
  }
  if (threadIdx.x == 0) atomicAdd(out, -red[0]);
}

// ---------------------------------------------------------------- launch
extern "C" void kernel_launch(void* const* d_in, const int* in_sizes, int n_in,
                              void* d_out, int out_size, void* d_ws, size_t ws_size,
                              hipStream_t stream) {
  const float* pc1 = (const float*)d_in[0];   // [16,30000,3] f32
  const float* pc2 = (const float*)d_in[1];   // [16,30000,3] f32
  const float* rot = (const float*)d_in[2];   // [2,3] f32
  float* out  = (float*)d_out;                // scalar f32
  float* imgs = (float*)d_ws;                 // 64 * 4096 f32 = 1 MiB

  init_kernel<<<256, 256, 0, stream>>>(imgs, out);
  splat_kernel<<<NIMG * SPLIT, 256, 0, stream>>>(pc1, pc2, rot, imgs);
  norm_bce_kernel<<<32, 256, 0, stream>>>(imgs, out);
}